// LineSegmentLoss_61134564492047
// MI455X (gfx1250) — compile-verified
//
#include <hip/hip_runtime.h>

// ---------------------------------------------------------------------------
// LineSegmentLoss fused reduction for MI455X (gfx1250, wave32)
// preds/gts: (32, 16, 256, 256) fp32.  Output: 1 scalar.
// Purely HBM-bound: 268 MB single pass -> ~11.5 us floor @ 23.3 TB/s.
// Cross-lane reduction done with V_WMMA_F32_16X16X4_F32 (ones-vector trick);
// the two wave-half partials are merged through LDS (plain stores, single
// exec-mask toggle, no atomics/shuffles in the epilogue).
// ---------------------------------------------------------------------------

typedef float v2f __attribute__((ext_vector_type(2)));
typedef float v4f __attribute__((ext_vector_type(4)));
typedef float v8f __attribute__((ext_vector_type(8)));

#define NB 32
#define NC 16
#define HW 65536                  // 256*256
#define BSTRIDE (NC * HW)         // 1048576 elements per batch
#define TOTPIX (NB * HW)          // 2097152 pixels
#define NGROUP (TOTPIX / 4)       // 524288 float4 pixel-groups
#define NACC 22
#define NCHUNK 16                 // 8 waves * 2 half-wave partials

// accumulator layout in d_ws (floats):
//  0.. 2  line_seg (c15): pos_sum, neg_sum, pos_cnt
//  3.. 5  junc_seg (c14)
//  6.. 8  sol_center (c0)
//  9..11  tp_center (c7)
// 12..13  sol_disp (c1..4): sum, cnt
// 14..15  tp_disp  (c8..11): sum, cnt
// 16..18  sol len/angle (c5,c6): len_sum, ang_sum, cnt
// 19..21  tp  len/angle (c12,c13)

__device__ __forceinline__ v4f ldnt(const float* p) {
    // 16B-aligned streaming load -> global_load_b128 with NT temporal hint
    return __builtin_nontemporal_load((const v4f*)p);
}

__device__ __forceinline__ float bce_logits(float x, float t) {
    // max(x,0) - x*t + log1p(exp(-|x|))
    return fmaxf(x, 0.0f) - x * t + __logf(1.0f + __expf(-fabsf(x)));
}

__device__ __forceinline__ float sl1(float d) {
    float a = fabsf(d);
    return (a < 1.0f) ? 0.5f * a * a : a - 0.5f;
}

__device__ __forceinline__ float sigmoidf(float x) {
    return 1.0f / (1.0f + __expf(-x));
}

// ---------------------------------------------------------------------------
// Half-wave sum via V_WMMA_F32_16X16X4_F32.
// A-matrix 16x4 f32 layout (ISA 7.12.2): VGPR0 lanes0-15 -> K=0 row M=lane,
// lanes16-31 -> K=2 row M=lane-16; VGPR1 -> K=1 / K=3.
// With A = {v, 0} and B = all-ones (layout-invariant): D[m][n] = v_m + v_{m+16}.
// C/D layout: lane j<16 holds rows M=0..7 in its 8 D VGPRs, lane j>=16 rows
// M=8..15, so per-lane sum of the 8 D regs gives:
//   lanes  0..15 : S_A = sum_{m=0..7} (v_m + v_{m+16})
//   lanes 16..31 : S_B = sum_{m=8..15}(v_m + v_{m+16})
// and S_A + S_B = full 32-lane sum (merged later through LDS).
// Requires EXEC all-ones: called only from the fully-converged epilogue.
// ---------------------------------------------------------------------------
__device__ __forceinline__ float wave_half_sum(float v) {
    v2f a;  a.x = v;    a.y = 0.0f;
    v2f b;  b.x = 1.0f; b.y = 1.0f;
    v8f c = {};
    v8f d = __builtin_amdgcn_wmma_f32_16x16x4_f32(
        /*neg_a=*/false, a, /*neg_b=*/false, b,
        /*c_mod=*/(short)0, c, /*reuse_a=*/false, /*reuse_b=*/false);
    return ((d[0] + d[1]) + (d[2] + d[3])) + ((d[4] + d[5]) + (d[6] + d[7]));
}

__global__ void __launch_bounds__(256)
zero_ws_kernel(float* __restrict__ ws) {
    if (threadIdx.x < NACC) ws[threadIdx.x] = 0.0f;
}

__global__ void __launch_bounds__(256)
loss_partial_kernel(const float* __restrict__ preds,
                    const float* __restrict__ gts,
                    float* __restrict__ ws) {
    __shared__ float s_part[NCHUNK * NACC];   // [wave-half chunk][acc]
    const int tid = threadIdx.x;

    float acc[NACC];
#pragma unroll
    for (int i = 0; i < NACC; ++i) acc[i] = 0.0f;

    const int stride = gridDim.x * blockDim.x;

    for (int g = blockIdx.x * blockDim.x + tid; g < NGROUP; g += stride) {
        const int p  = g << 2;            // pixel index
        const int b  = p >> 16;           // / HW
        const int hw = p & (HW - 1);
        const float* pb = preds + (size_t)b * BSTRIDE + hw;
        const float* gb = gts   + (size_t)b * BSTRIDE + hw;

        // Prefetch next grid-stride iteration's cache lines (global_prefetch_b8)
        const int gn = g + stride;
        if (gn < NGROUP) {
            const int pn  = gn << 2;
            const int bn  = pn >> 16;
            const int hwn = pn & (HW - 1);
            __builtin_prefetch(preds + (size_t)bn * BSTRIDE + hwn, 0, 0);
            __builtin_prefetch(gts   + (size_t)bn * BSTRIDE + hwn, 0, 0);
        }

        // ---- weighted-BCE heads: channel c -> acc[a0..a0+2] ----
        auto bce_head = [&](int c, int a0) {
            v4f x = ldnt(pb + c * HW);
            v4f t = ldnt(gb + c * HW);
#pragma unroll
            for (int i = 0; i < 4; ++i) {
                float l = bce_logits(x[i], t[i]);
                float m = (t[i] != 0.0f) ? 1.0f : 0.0f;
                float lm = l * m;
                acc[a0 + 0] += lm;
                acc[a0 + 1] += l - lm;
                acc[a0 + 2] += m;
            }
        };

        // ---- displacement heads: channels c0..c0+3 -> acc[a0..a0+1] ----
        auto disp_head = [&](int c0, int a0) {
            v4f p1 = ldnt(pb + (c0 + 0) * HW);
            v4f p2 = ldnt(pb + (c0 + 1) * HW);
            v4f p3 = ldnt(pb + (c0 + 2) * HW);
            v4f p4 = ldnt(pb + (c0 + 3) * HW);
            v4f g1 = ldnt(gb + (c0 + 0) * HW);
            v4f g2 = ldnt(gb + (c0 + 1) * HW);
            v4f g3 = ldnt(gb + (c0 + 2) * HW);
            v4f g4 = ldnt(gb + (c0 + 3) * HW);
#pragma unroll
            for (int i = 0; i < 4; ++i) {
                float pv = fabsf(g1[i]) + fabsf(g2[i]) + fabsf(g3[i]) + fabsf(g4[i]);
                float m  = (pv != 0.0f) ? 1.0f : 0.0f;
                float l1 = sl1(p1[i] - g1[i]) + sl1(p2[i] - g2[i]) +
                           sl1(p3[i] - g3[i]) + sl1(p4[i] - g4[i]);
                // pred_swap = [c0+2, c0+3, c0+0, c0+1]
                float l2 = sl1(p3[i] - g1[i]) + sl1(p4[i] - g2[i]) +
                           sl1(p1[i] - g3[i]) + sl1(p2[i] - g4[i]);
                acc[a0 + 0] += m * fminf(l1, l2);
                acc[a0 + 1] += m;
            }
        };

        // ---- len/angle heads: channels cl, ca -> acc[a0..a0+2] ----
        auto la_head = [&](int cl, int ca, int a0) {
            v4f pl = ldnt(pb + cl * HW);
            v4f pa = ldnt(pb + ca * HW);
            v4f gl = ldnt(gb + cl * HW);
            v4f ga = ldnt(gb + ca * HW);
#pragma unroll
            for (int i = 0; i < 4; ++i) {
                float m  = (gl[i] != 0.0f) ? 1.0f : 0.0f;
                float sl = sigmoidf(pl[i]);
                float sa = sigmoidf(pa[i]);
                acc[a0 + 0] += sl1(sl - gl[i]) * m;
                acc[a0 + 1] += sl1(sa - ga[i]) * m;
                acc[a0 + 2] += m;
            }
        };

        bce_head(15, 0);    // line_seg  (w 1,1)
        bce_head(14, 3);    // junc_seg  (w 1,30)
        bce_head(0, 6);     // sol_center(w 1,30)
        bce_head(7, 9);     // tp_center (w 1,30)
        disp_head(1, 12);   // sol_disp
        disp_head(8, 14);   // tp_disp
        la_head(5, 6, 16);  // sol len/angle
        la_head(12, 13, 19);// tp len/angle
    }

    // ---- epilogue ----
    // 1) All lanes converged (WMMA needs EXEC all-ones): reduce each
    //    accumulator to its wave-half partial with one WMMA each.
#pragma unroll
    for (int i = 0; i < NACC; ++i)
        acc[i] = wave_half_sum(acc[i]);            // v_wmma_f32_16x16x4_f32

    // 2) Single exec-mask toggle: lanes 0 and 16 of each wave store their
    //    22 half-partials contiguously (ds_store_b32 clause).
    const int lane  = tid & 31;
    const int chunk = (tid >> 5 << 1) | (lane >> 4);   // wave*2 + half
    if ((lane & 15) == 0) {
#pragma unroll
        for (int i = 0; i < NACC; ++i)
            s_part[chunk * NACC + i] = acc[i];
    }
    __syncthreads();

    // 3) Threads 0..21 fold the 16 chunks and emit one divergent-address
    //    global_atomic_add_f32 each (no atomic-optimizer expansion).
    if (tid < NACC) {
        float t = 0.0f;
#pragma unroll
        for (int k = 0; k < NCHUNK; ++k) t += s_part[k * NACC + tid];
        atomicAdd(&ws[tid], t);
    }
}

__global__ void finalize_kernel(const float* __restrict__ ws,
                                float* __restrict__ out) {
    const float TOT = (float)TOTPIX;
    // weighted BCE: pos_sum/pos_cnt * pw + neg_sum/(TOT-pos_cnt) * nw
    float line = 1.0f  * ws[0] / ws[2]  + 1.0f  * ws[1]  / (TOT - ws[2]);
    float junc = 1.0f  * ws[3] / ws[5]  + 30.0f * ws[4]  / (TOT - ws[5]);
    float solc = 1.0f  * ws[6] / ws[8]  + 30.0f * ws[7]  / (TOT - ws[8]);
    float tpc  = 1.0f  * ws[9] / ws[11] + 30.0f * ws[10] / (TOT - ws[11]);
    float sold = ws[12] / ws[13];
    float tpd  = ws[14] / ws[15];
    float soll = ws[16] / ws[18];
    float sola = ws[17] / ws[18];
    float tpl  = ws[19] / ws[21];
    float tpa  = ws[20] / ws[21];
    // total = 10*tp_center + tp_disp + tp_len + tp_angle
    //       + sol_center + sol_disp + sol_len + sol_angle + line + junc
    out[0] = 10.0f * tpc + tpd + tpl + tpa + solc + sold + soll + sola + line + junc;
}

extern "C" void kernel_launch(void* const* d_in, const int* in_sizes, int n_in,
                              void* d_out, int out_size, void* d_ws, size_t ws_size,
                              hipStream_t stream) {
    const float* preds = (const float*)d_in[0];
    const float* gts   = (const float*)d_in[1];
    // d_in[2], d_in[3] (tp/sol gt lines) are unused by the reference.
    float* ws  = (float*)d_ws;   // 22 floats of accumulators
    float* out = (float*)d_out;

    zero_ws_kernel<<<1, 32, 0, stream>>>(ws);

    // 1024 blocks * 256 threads (8 waves/block) -> 2 float4-groups per thread,
    // so the prefetch of iteration g+stride overlaps iteration g's compute.
    loss_partial_kernel<<<1024, 256, 0, stream>>>(preds, gts, ws);

    finalize_kernel<<<1, 1, 0, stream>>>(ws, out);
}